// EGraphSAGE_59021440582265
// MI455X (gfx1250) — compile-verified
//
#include <hip/hip_runtime.h>

typedef __attribute__((ext_vector_type(2))) float v2f;
typedef __attribute__((ext_vector_type(8))) float v8f;

#define NNODES 100000
#define NEDGES 1600000
#define F0 128
#define H1 128
#define H2 64

// ---------------- zero fill (float4) ----------------
__global__ void k_zero4(float* __restrict__ p, long n /* multiple of 4 */) {
    long i4 = ((long)blockIdx.x * blockDim.x + threadIdx.x) * 4;
    if (i4 < n) {
        float4 z = make_float4(0.f, 0.f, 0.f, 0.f);
        *(float4*)(p + i4) = z;
    }
}

// ------- layer-1 scatter: edge_attr -> agg1 (+ counts, shared by both layers) -------
// thread = (edge, 4-float chunk); 32 chunks cover 128 dims; coalesced float4 reads.
__global__ void k_scatter1(const float* __restrict__ ea, const int* __restrict__ src,
                           float* __restrict__ agg, float* __restrict__ cnt) {
    long tid = (long)blockIdx.x * blockDim.x + threadIdx.x;
    int e = (int)(tid >> 5);
    int c = (int)(tid & 31);
    if (e >= NEDGES) return;
    int s = src[e];
    float4 v = *(const float4*)(ea + (long)e * F0 + c * 4);
    float* d = agg + (long)s * F0 + c * 4;
    atomicAdd(d + 0, v.x);
    atomicAdd(d + 1, v.y);
    atomicAdd(d + 2, v.z);
    atomicAdd(d + 3, v.w);
    if (c == 0) atomicAdd(cnt + s, 1.0f);
}

// ------- fused layer-2 scatter: x1[e]=(ne1[src]+ne1[dst])/2 scattered into agg2 -------
// avoids ever materializing the 819MB x1 buffer; ne1 (51MB) is L2-resident.
__global__ void k_scatter2(const float* __restrict__ ne1, const int* __restrict__ src,
                           const int* __restrict__ dst, float* __restrict__ agg2) {
    long tid = (long)blockIdx.x * blockDim.x + threadIdx.x;
    int e = (int)(tid >> 5);
    int c = (int)(tid & 31);
    if (e >= NEDGES) return;
    int s = src[e], d = dst[e];
    float4 a = *(const float4*)(ne1 + (long)s * H1 + c * 4);
    float4 b = *(const float4*)(ne1 + (long)d * H1 + c * 4);
    float* o = agg2 + (long)s * H1 + c * 4;
    atomicAdd(o + 0, 0.5f * (a.x + b.x));
    atomicAdd(o + 1, 0.5f * (a.y + b.y));
    atomicAdd(o + 2, 0.5f * (a.z + b.z));
    atomicAdd(o + 3, 0.5f * (a.w + b.w));
}

// ------- mean finalize (in place): agg -> ms = 1 + 0.5 * (cnt>0 ? agg/cnt : 0) -------
__global__ void k_finalize_mean(float* __restrict__ agg, const float* __restrict__ cnt,
                                int log2C, long total) {
    long i = (long)blockIdx.x * blockDim.x + threadIdx.x;
    if (i >= total) return;
    float c = cnt[i >> log2C];
    float m = (c > 0.f) ? (agg[i] / c) : 0.f;
    agg[i] = 1.0f + 0.5f * m;
}

// ------- fp32 WMMA GEMM + bias + sigmoid: out[M x N] = sigmoid(A[M x K] @ W[K x N] + b) -------
// One wave32 per 16x16 output tile. M=100000 = 6250*16, N in {128,64}, K=128: no predication,
// EXEC is all-ones as V_WMMA requires.
__global__ void __launch_bounds__(32)
k_wmma_gemm_sigmoid(const float* __restrict__ A, const float* __restrict__ W,
                    const float* __restrict__ bias, float* __restrict__ out,
                    int K, int N) {
    const int tm = blockIdx.x * 16;
    const int tn = blockIdx.y * 16;
    const int lane = threadIdx.x;      // 0..31
    const int hi = lane >> 4;          // half-wave select
    const int mr = lane & 15;

    const float* arow = A + (long)(tm + mr) * K + hi * 2;        // A[mr][k0+2hi..]
    const float* wcol = W + (long)(hi * 2) * N + tn + mr;        // W[k0+2hi..][tn+mr]

    v8f acc = {};
    for (int k = 0; k < K; k += 4) {
        v2f a; a.x = arow[0]; a.y = arow[1];
        v2f b; b.x = wcol[0]; b.y = wcol[N];
        acc = __builtin_amdgcn_wmma_f32_16x16x4_f32(
            /*neg_a=*/false, a, /*neg_b=*/false, b,
            /*c_mod=*/(short)0, acc, /*reuse_a=*/false, /*reuse_b=*/false);
        arow += 4;
        wcol += (long)4 * N;
    }

    const int col = tn + mr;
    const float bv = bias[col];
#pragma unroll
    for (int r = 0; r < 8; ++r) {
        int m = tm + r + hi * 8;                 // D layout: vgpr r -> row r (+8 for lanes 16-31)
        float v = acc[r] + bv;
        out[(long)m * N + col] = 1.0f / (1.0f + __expf(-v));
    }
}

// ------- final fused gather + head: emb[e]=(ne2[src]+ne2[dst])/2 ; logit[e]=emb.Wf+bf -------
// one wave32 per edge, 2 of 64 dims per lane, shfl_xor butterfly reduction for the dot.
__global__ void k_edge_final(const float* __restrict__ ne2, const int* __restrict__ src,
                             const int* __restrict__ dst, const float* __restrict__ Wf,
                             const float* __restrict__ bf, float* __restrict__ logits,
                             float* __restrict__ emb) {
    int wid = blockIdx.x * (blockDim.x >> 5) + (threadIdx.x >> 5);
    int lane = threadIdx.x & 31;
    if (wid >= NEDGES) return;
    int s = src[wid], d = dst[wid];
    float2 a = *(const float2*)(ne2 + (long)s * H2 + lane * 2);
    float2 b = *(const float2*)(ne2 + (long)d * H2 + lane * 2);
    float2 e;
    e.x = 0.5f * (a.x + b.x);
    e.y = 0.5f * (a.y + b.y);
    *(float2*)(emb + (long)wid * H2 + lane * 2) = e;
    float p = e.x * Wf[lane * 2] + e.y * Wf[lane * 2 + 1];
#pragma unroll
    for (int off = 16; off > 0; off >>= 1) p += __shfl_xor(p, off, 32);
    if (lane == 0) logits[wid] = p + bf[0];
}

extern "C" void kernel_launch(void* const* d_in, const int* in_sizes, int n_in,
                              void* d_out, int out_size, void* d_ws, size_t ws_size,
                              hipStream_t stream) {
    const float* edge_attr = (const float*)d_in[0];
    const int*   eidx      = (const int*)d_in[1];   // (2, NEDGES)
    const float* W1        = (const float*)d_in[3];
    const float* b1        = (const float*)d_in[4];
    const float* W2        = (const float*)d_in[5];
    const float* b2        = (const float*)d_in[6];
    const float* Wf        = (const float*)d_in[7];
    const float* bf        = (const float*)d_in[8];
    const int* src = eidx;
    const int* dst = eidx + NEDGES;

    // workspace layout (floats)
    float* ws   = (float*)d_ws;
    float* cnt  = ws;                              // NNODES
    float* agg1 = cnt + NNODES;                    // NNODES*F0   (becomes ms1 in place)
    float* ne1  = agg1 + (long)NNODES * F0;        // NNODES*H1
    float* agg2 = ne1 + (long)NNODES * H1;         // NNODES*H1   (becomes ms2 in place)
    float* ne2  = agg2 + (long)NNODES * H1;        // NNODES*H2

    float* logits = (float*)d_out;                 // NEDGES
    float* emb    = logits + NEDGES;               // NEDGES*H2

    const int BLK = 256;

    // zero cnt+agg1 (contiguous) and agg2
    long zn1 = (long)NNODES * (1 + F0);
    long zn2 = (long)NNODES * H1;
    k_zero4<<<(int)((zn1 / 4 + BLK - 1) / BLK), BLK, 0, stream>>>(cnt, zn1);
    k_zero4<<<(int)((zn2 / 4 + BLK - 1) / BLK), BLK, 0, stream>>>(agg2, zn2);

    // layer 1
    long t1 = (long)NEDGES * 32;
    k_scatter1<<<(int)((t1 + BLK - 1) / BLK), BLK, 0, stream>>>(edge_attr, src, agg1, cnt);
    long n1 = (long)NNODES * F0;
    k_finalize_mean<<<(int)((n1 + BLK - 1) / BLK), BLK, 0, stream>>>(agg1, cnt, 7, n1);
    k_wmma_gemm_sigmoid<<<dim3(NNODES / 16, H1 / 16), 32, 0, stream>>>(agg1, W1, b1, ne1, F0, H1);

    // layer 2 (fused edge-update + scatter)
    k_scatter2<<<(int)((t1 + BLK - 1) / BLK), BLK, 0, stream>>>(ne1, src, dst, agg2);
    long n2 = (long)NNODES * H1;
    k_finalize_mean<<<(int)((n2 + BLK - 1) / BLK), BLK, 0, stream>>>(agg2, cnt, 7, n2);
    k_wmma_gemm_sigmoid<<<dim3(NNODES / 16, H2 / 16), 32, 0, stream>>>(agg2, W2, b2, ne2, H1, H2);

    // final gather + linear head
    k_edge_final<<<NEDGES / 8, BLK, 0, stream>>>(ne2, src, dst, Wf, bf, logits, emb);
}